// FeatureRegression_29884382445647
// MI455X (gfx1250) — compile-verified
//
#include <hip/hip_runtime.h>
#include <hip/hip_bf16.h>

typedef float v2f __attribute__((ext_vector_type(2)));
typedef float v8f __attribute__((ext_vector_type(8)));

#define BN_EPS 1e-5f

__device__ __forceinline__ v8f wmma_f32(v2f a, v2f b, v8f c) {
  // 8 args: (neg_a, A, neg_b, B, c_mod, C, reuse_a, reuse_b)
  return __builtin_amdgcn_wmma_f32_16x16x4_f32(false, a, false, b, (short)0, c,
                                               false, false);
}

// ---------------------------------------------------------------------------
// Prep 1: xt[pos][c][b] = x[b][c][pos], c padded 225 -> 228 with zeros.
// grid: (ceil(225/16)=15, 256/16=16, 228), block (16,16)
// ---------------------------------------------------------------------------
__global__ __launch_bounds__(256) void transpose_pad_kernel(
    const float* __restrict__ x, float* __restrict__ xt) {
  __shared__ float T[16][17];
  const int c = blockIdx.z;            // 0..227
  const int pos0 = blockIdx.x * 16;
  const int b0 = blockIdx.y * 16;
  const int tx = threadIdx.x, ty = threadIdx.y;
  float v = 0.0f;
  const int pos = pos0 + tx;
  if (c < 225 && pos < 225)
    v = x[((long)(b0 + ty) * 225 + c) * 225 + pos];
  T[ty][tx] = v;
  __syncthreads();
  const int posw = pos0 + ty;
  if (posw < 225)
    xt[((long)posw * 228 + c) * 256 + b0 + tx] = T[tx][ty];
}

// ---------------------------------------------------------------------------
// Prep 2: wprime[pos][o][cpad] = w1[o][(c/15 + 14 - i)*29 + (c%15 + 14 - j)]
// (the align gather folded into the weight), zero for cpad >= 225.
// ---------------------------------------------------------------------------
__global__ __launch_bounds__(256) void build_wprime_kernel(
    const float* __restrict__ w1, float* __restrict__ wp) {
  const long idx = (long)blockIdx.x * 256 + threadIdx.x;  // 225*128*228 total
  const int cpad = (int)(idx % 228);
  const long t2 = idx / 228;
  const int o = (int)(t2 % 128);
  const int pos = (int)(t2 / 128);
  float v = 0.0f;
  if (cpad < 225) {
    const int i = pos / 15, j = pos % 15;
    const int ih = cpad / 15, iw = cpad % 15;
    const int e = (ih + 14 - i) * 29 + (iw + 14 - j);
    v = w1[(long)o * 841 + e];
  }
  wp[idx] = v;
}

// ---------------------------------------------------------------------------
// Prep 3: concat buffer [136][20736]: rows 0..4 = broadcast weight(5,9,9),
// rows 133..135 = 0 (K padding).  Rows 5..132 filled later by conv7 kernel.
// ---------------------------------------------------------------------------
__global__ __launch_bounds__(256) void catfill_kernel(
    const float* __restrict__ weight, float* __restrict__ catbuf) {
  const int idx = blockIdx.x * 256 + threadIdx.x;  // 8*20736 total
  const int n = idx % 20736;
  const int rr = idx / 20736;
  if (rr < 5) {
    const int p = n / 256;
    catbuf[(long)rr * 20736 + n] = weight[rr * 81 + p];
  } else {
    catbuf[(long)(128 + rr) * 20736 + n] = 0.0f;  // rows 133..135
  }
}

// ---------------------------------------------------------------------------
// Prep 4: pad w3 (128x133) -> w3p (128x136), zero columns 133..135.
// ---------------------------------------------------------------------------
__global__ __launch_bounds__(256) void w3pad_kernel(
    const float* __restrict__ w3, float* __restrict__ w3p) {
  const int idx = blockIdx.x * 256 + threadIdx.x;  // 128*136 = 17408 total
  if (idx >= 128 * 136) return;
  const int col = idx % 136;
  const int row = idx / 136;
  w3p[idx] = (col < 133) ? w3[row * 133 + col] : 0.0f;
}

// ---------------------------------------------------------------------------
// Generic WMMA f32 GEMM: Y[m,n] = epilogue( sum_k A[m,k] * X[k,n] )
// epilogue = optional ReLU( BN( acc + bias[m] ) )
// block = 256 threads (8 waves); wave w owns M-tile [16w,16w+16) x 64 N cols.
// grid.x = N/64, grid.z = batch (per-batch strides in elements).
// Kpad multiple of 4; A has Kpad columns, X has Kpad rows (zero padded).
// Double-buffered LDS staging: global prefetch overlaps the WMMA chain.
// ---------------------------------------------------------------------------
__global__ __launch_bounds__(256) void gemm_bn_relu_kernel(
    const float* __restrict__ A, int lda, long a_bstride,
    const float* __restrict__ X, int ldx, long x_bstride,
    float* __restrict__ Y, int ldy, long y_bstride,
    const float* __restrict__ bias, const float* __restrict__ bn,
    int M, int Kpad, int relu) {
  __shared__ float sX[2][16][72];  // stride 72: half-waves on disjoint banks
  const int t = threadIdx.x;
  const int lane = t & 31, wv = t >> 5;
  const int lm = lane & 15, hi = lane >> 4;
  const int n0 = blockIdx.x * 64;
  const long zb = blockIdx.z;
  A += zb * a_bstride;
  X += zb * x_bstride;
  Y += zb * y_bstride;
  const int m0 = wv * 16;
  const bool active = (m0 < M);

  v8f acc[4];
#pragma unroll
  for (int nb = 0; nb < 4; ++nb)
#pragma unroll
    for (int e = 0; e < 8; ++e) acc[nb][e] = 0.0f;

  const int srow = t >> 4;          // 0..15
  const int scol = (t & 15) * 4;    // 0..60
  const float* Arow = A + (long)(m0 + lm) * lda;
  const int nch = (Kpad + 15) >> 4;

  // Prologue: stage chunk 0 into buffer 0.
  {
    float4 v0 = make_float4(0.f, 0.f, 0.f, 0.f);
    if (srow < Kpad) v0 = *(const float4*)(X + (long)srow * ldx + n0 + scol);
    sX[0][srow][scol + 0] = v0.x;
    sX[0][srow][scol + 1] = v0.y;
    sX[0][srow][scol + 2] = v0.z;
    sX[0][srow][scol + 3] = v0.w;
  }
  __syncthreads();

  for (int ic = 0; ic < nch; ++ic) {
    const int k0 = ic << 4;
    const int cur = ic & 1, nxt = cur ^ 1;
    const bool havenext = (ic + 1 < nch);
    float4 pre;
    if (havenext) {
      const int kk = k0 + 16 + srow;
      pre = (kk < Kpad) ? *(const float4*)(X + (long)kk * ldx + n0 + scol)
                        : make_float4(0.f, 0.f, 0.f, 0.f);
    }
    if (active) {
      const int ksteps = min(16, Kpad - k0);  // multiple of 4
      for (int s = 0; s < ksteps; s += 4) {
        const int kl = s + hi * 2;
        const v2f a = *(const v2f*)(Arow + k0 + kl);  // global_load_b64
#pragma unroll
        for (int nb = 0; nb < 4; ++nb) {
          v2f b;
          b[0] = sX[cur][kl][nb * 16 + lm];
          b[1] = sX[cur][kl + 1][nb * 16 + lm];
          acc[nb] = wmma_f32(a, b, acc[nb]);
        }
      }
    }
    if (havenext) {
      sX[nxt][srow][scol + 0] = pre.x;
      sX[nxt][srow][scol + 1] = pre.y;
      sX[nxt][srow][scol + 2] = pre.z;
      sX[nxt][srow][scol + 3] = pre.w;
    }
    __syncthreads();
  }

  if (active) {
#pragma unroll
    for (int v = 0; v < 8; ++v) {
      const int m = m0 + hi * 8 + v;
      const float scale = bn[m] * rsqrtf(bn[3 * M + m] + BN_EPS);
      const float shift = bn[M + m] + (bias[m] - bn[2 * M + m]) * scale;
#pragma unroll
      for (int nb = 0; nb < 4; ++nb) {
        const int n = n0 + nb * 16 + lm;
        float val = acc[nb][v] * scale + shift;
        if (relu) val = fmaxf(val, 0.0f);
        Y[(long)m * ldy + n] = val;
      }
    }
  }
}

// ---------------------------------------------------------------------------
// 7x7 VALID conv as implicit-im2col WMMA GEMM (double-buffered).
// y1: [c=128][pos=225][b=256] (stage-1 activations).
// out rows written to catbuf rows 5..132: out[(5+o)*20736 + p*256 + b].
// grid: x = 4 (64-wide batch strip), z = 81 (output position p = ph*9+pw).
// K = 128*49 = 6272, k -> (c = k/49, kh = (k%49)/7, kw = k%7).
// ---------------------------------------------------------------------------
__global__ __launch_bounds__(256) void conv7_gemm_kernel(
    const float* __restrict__ y1, const float* __restrict__ w2,
    const float* __restrict__ b2, const float* __restrict__ bn2,
    float* __restrict__ catbuf) {
  __shared__ float sX[2][16][72];
  const int t = threadIdx.x;
  const int lane = t & 31, wv = t >> 5;
  const int lm = lane & 15, hi = lane >> 4;
  const int p = blockIdx.z;  // 0..80
  const int ph = p / 9, pw = p % 9;
  const int b0 = blockIdx.x * 64;
  const int m0 = wv * 16;

  v8f acc[4];
#pragma unroll
  for (int nb = 0; nb < 4; ++nb)
#pragma unroll
    for (int e = 0; e < 8; ++e) acc[nb][e] = 0.0f;

  const int srow = t >> 4;
  const int scol = (t & 15) * 4;
  const float* Arow = w2 + (long)(m0 + lm) * 6272;

  // Prologue: stage chunk 0.
  {
    const int c = srow / 49, r = srow % 49;  // srow < 49 here, so c==0
    const int sp = (ph + r / 7) * 15 + (pw + r % 7);
    const float4 v0 =
        *(const float4*)(y1 + ((long)c * 225 + sp) * 256 + b0 + scol);
    sX[0][srow][scol + 0] = v0.x;
    sX[0][srow][scol + 1] = v0.y;
    sX[0][srow][scol + 2] = v0.z;
    sX[0][srow][scol + 3] = v0.w;
  }
  __syncthreads();

  for (int ic = 0; ic < 392; ++ic) {  // 6272 / 16
    const int k0 = ic << 4;
    const int cur = ic & 1, nxt = cur ^ 1;
    const bool havenext = (ic + 1 < 392);
    float4 pre;
    if (havenext) {
      const int k = k0 + 16 + srow;
      const int c = k / 49, r = k % 49;
      const int sp = (ph + r / 7) * 15 + (pw + r % 7);
      pre = *(const float4*)(y1 + ((long)c * 225 + sp) * 256 + b0 + scol);
    }
#pragma unroll
    for (int s = 0; s < 16; s += 4) {
      const int kl = s + hi * 2;
      const v2f a = *(const v2f*)(Arow + k0 + kl);
#pragma unroll
      for (int nb = 0; nb < 4; ++nb) {
        v2f b;
        b[0] = sX[cur][kl][nb * 16 + lm];
        b[1] = sX[cur][kl + 1][nb * 16 + lm];
        acc[nb] = wmma_f32(a, b, acc[nb]);
      }
    }
    if (havenext) {
      sX[nxt][srow][scol + 0] = pre.x;
      sX[nxt][srow][scol + 1] = pre.y;
      sX[nxt][srow][scol + 2] = pre.z;
      sX[nxt][srow][scol + 3] = pre.w;
    }
    __syncthreads();
  }

#pragma unroll
  for (int v = 0; v < 8; ++v) {
    const int m = m0 + hi * 8 + v;
    const float scale = bn2[m] * rsqrtf(bn2[384 + m] + BN_EPS);
    const float shift = bn2[128 + m] + (b2[m] - bn2[256 + m]) * scale;
#pragma unroll
    for (int nb = 0; nb < 4; ++nb) {
      const int n = p * 256 + b0 + nb * 16 + lm;
      const float val = fmaxf(acc[nb][v] * scale + shift, 0.0f);
      catbuf[(long)(5 + m) * 20736 + n] = val;
    }
  }
}

// ---------------------------------------------------------------------------
// Attention score: a2[n] = ba2 + sum_{o<64} wa2[o] * a1[o][n]
// ---------------------------------------------------------------------------
__global__ __launch_bounds__(256) void att_score_kernel(
    const float* __restrict__ a1, const float* __restrict__ wa2,
    const float* __restrict__ ba2, float* __restrict__ a2) {
  const int n = blockIdx.x * 256 + threadIdx.x;  // 20736 total
  float s = ba2[0];
  for (int o = 0; o < 64; ++o) s += wa2[o] * a1[(long)o * 20736 + n];
  a2[n] = s;
}

// ---------------------------------------------------------------------------
// Per-batch softmax over 81 positions, attention pooling over merge, 6-wide FC.
// grid = 256 (batch), block = 128.
// ---------------------------------------------------------------------------
__global__ __launch_bounds__(128) void softmax_pool_fc_kernel(
    const float* __restrict__ a2, const float* __restrict__ merge,
    const float* __restrict__ wl, const float* __restrict__ bl,
    float* __restrict__ out) {
  __shared__ float satt[81];
  __shared__ float sp[128];
  const int b = blockIdx.x;
  const int t = threadIdx.x;
  if (t < 81) satt[t] = a2[t * 256 + b];
  __syncthreads();
  if (t == 0) {
    float mx = satt[0];
    for (int p = 1; p < 81; ++p) mx = fmaxf(mx, satt[p]);
    float sum = 0.0f;
    for (int p = 0; p < 81; ++p) {
      const float e = expf(satt[p] - mx);
      satt[p] = e;
      sum += e;
    }
    const float inv = 1.0f / sum;
    for (int p = 0; p < 81; ++p) satt[p] *= inv;
  }
  __syncthreads();
  float pooled = 0.0f;
  const float* mrow = merge + (long)t * 20736 + b;
  for (int p = 0; p < 81; ++p) pooled += satt[p] * mrow[p * 256];
  sp[t] = pooled;
  __syncthreads();
  if (t < 6) {
    float s = bl[t];
    for (int o = 0; o < 128; ++o) s += sp[o] * wl[t * 128 + o];
    out[b * 6 + t] = s;
  }
}

// ---------------------------------------------------------------------------
// Host launcher
// ---------------------------------------------------------------------------
extern "C" void kernel_launch(void* const* d_in, const int* in_sizes, int n_in,
                              void* d_out, int out_size, void* d_ws,
                              size_t ws_size, hipStream_t stream) {
  const float* x = (const float*)d_in[0];       // (256,225,15,15)
  const float* w1 = (const float*)d_in[1];      // (128,841)
  const float* b1 = (const float*)d_in[2];      // (128)
  const float* bn1 = (const float*)d_in[3];     // (4,128)
  const float* w2 = (const float*)d_in[4];      // (128,128,7,7)
  const float* b2 = (const float*)d_in[5];
  const float* bn2 = (const float*)d_in[6];
  const float* weight = (const float*)d_in[7];  // (1,5,9,9)
  const float* w3 = (const float*)d_in[8];      // (128,133)
  const float* b3 = (const float*)d_in[9];
  const float* bn3 = (const float*)d_in[10];
  const float* w4 = (const float*)d_in[11];     // (128,128)
  const float* b4 = (const float*)d_in[12];
  const float* bn4 = (const float*)d_in[13];
  const float* wa1 = (const float*)d_in[14];    // (64,128)
  const float* ba1 = (const float*)d_in[15];
  const float* bna = (const float*)d_in[16];    // (4,64)
  const float* wa2 = (const float*)d_in[17];    // (1,64)
  const float* ba2 = (const float*)d_in[18];
  const float* wl = (const float*)d_in[19];     // (6,128)
  const float* bl = (const float*)d_in[20];
  float* out = (float*)d_out;                   // (256,6)

  float* ws = (float*)d_ws;
  float* xt = ws;                                 // 225*228*256
  float* wprime = xt + (long)225 * 228 * 256;     // 225*128*228
  float* y1 = wprime + (long)225 * 128 * 228;     // 128*225*256
  float* catbuf = y1 + (long)128 * 225 * 256;     // 136*20736
  float* x3 = catbuf + (long)136 * 20736;         // 128*20736
  float* mergeb = x3 + (long)128 * 20736;         // 128*20736
  float* a1 = mergeb + (long)128 * 20736;         // 64*20736
  float* a2 = a1 + (long)64 * 20736;              // 20736
  float* w3p = a2 + 20736;                        // 128*136

  // Prep
  transpose_pad_kernel<<<dim3(15, 16, 228), dim3(16, 16), 0, stream>>>(x, xt);
  build_wprime_kernel<<<dim3(25650), dim3(256), 0, stream>>>(w1, wprime);
  catfill_kernel<<<dim3(648), dim3(256), 0, stream>>>(weight, catbuf);
  w3pad_kernel<<<dim3(68), dim3(256), 0, stream>>>(w3, w3p);

  // Stage 1: per-position align+conv1x1 GEMM, batched over 225 positions.
  gemm_bn_relu_kernel<<<dim3(4, 1, 225), dim3(256), 0, stream>>>(
      wprime, 228, (long)128 * 228, xt, 256, (long)228 * 256, y1, 57600,
      (long)256, b1, bn1, 128, 228, 1);

  // Stage 2: 7x7 conv implicit GEMM -> catbuf rows 5..132.
  conv7_gemm_kernel<<<dim3(4, 1, 81), dim3(256), 0, stream>>>(y1, w2, b2, bn2,
                                                              catbuf);

  // Stage 3: conv1x1 w3 (zero-padded to K=136) over concat(weight, y2).
  gemm_bn_relu_kernel<<<dim3(324, 1, 1), dim3(256), 0, stream>>>(
      w3p, 136, 0L, catbuf, 20736, 0L, x3, 20736, 0L, b3, bn3, 128, 136, 1);

  // Stage 4: conv1x1 w4 -> merge.
  gemm_bn_relu_kernel<<<dim3(324, 1, 1), dim3(256), 0, stream>>>(
      w4, 128, 0L, x3, 20736, 0L, mergeb, 20736, 0L, b4, bn4, 128, 128, 1);

  // Stage 5: attention conv1x1 wa1 on y2 (catbuf rows 5..132) -> a1.
  gemm_bn_relu_kernel<<<dim3(324, 1, 1), dim3(256), 0, stream>>>(
      wa1, 128, 0L, catbuf + (long)5 * 20736, 20736, 0L, a1, 20736, 0L, ba1,
      bna, 64, 128, 1);

  // Stage 6: attention score + softmax pooling + final linear.
  att_score_kernel<<<dim3(81), dim3(256), 0, stream>>>(a1, wa2, ba2, a2);
  softmax_pool_fc_kernel<<<dim3(256), dim3(128), 0, stream>>>(a2, mergeb, wl,
                                                              bl, out);
}